// Ensemble_postprocess_76656576299744
// MI455X (gfx1250) — compile-verified
//
#include <hip/hip_runtime.h>
#include <stdint.h>

#define NBOX 8700
#define NPAD 8704   // NBOX rounded up to multiple of 16 (and of 4 for b128 copies)
#define NYOLO 8400
#define NRT 300

typedef float v2f __attribute__((ext_vector_type(2)));
typedef float v8f __attribute__((ext_vector_type(8)));

// ---------------- ws layout (in floats from base) ----------------
// 0*NPAD  x1      (original order)
// 1*NPAD  y1
// 2*NPAD  x2
// 3*NPAD  y2
// 4*NPAD  score
// 5*NPAD  key     (score>=0.5 ? score : -1 ; pads = -2)
// 6*NPAD  area
// 7*NPAD  sbox4   (sorted, float4 xyxy, occupies 4*NPAD floats)
// 11*NPAD sscore  (sorted)
// 12*NPAD sarea   (sorted)
// 13*NPAD M       (int: number of valid boxes)
// total: 13*NPAD*4 + 4 bytes  ~= 452 KB

// ============ Kernel 1: preprocess (single block) ============
__global__ __launch_bounds__(1024) void prep_kernel(const float* __restrict__ yolo,
                                                    const float* __restrict__ rt,
                                                    float* __restrict__ ws) {
  __shared__ float red[1024];
  __shared__ int scount;
  const int tid = threadIdx.x;

  // max of rt conf (300 entries)
  float v = -1e30f;
  if (tid < NRT) v = rt[tid * 5 + 4];
  red[tid] = v;
  if (tid == 0) scount = 0;
  __syncthreads();
  for (int s = 512; s > 0; s >>= 1) {
    if (tid < s) red[tid] = fmaxf(red[tid], red[tid + s]);
    __syncthreads();
  }
  const float rmax = red[0];

  float* x1 = ws;
  float* y1 = ws + 1 * NPAD;
  float* x2 = ws + 2 * NPAD;
  float* y2 = ws + 3 * NPAD;
  float* sc = ws + 4 * NPAD;
  float* ky = ws + 5 * NPAD;
  float* ar = ws + 6 * NPAD;
  float4* sbox = (float4*)(ws + 7 * NPAD);
  float* ssc = ws + 11 * NPAD;
  float* sar = ws + 12 * NPAD;

  int cnt = 0;
  for (int i = tid; i < NPAD; i += 1024) {
    if (i < NBOX) {
      float cx, cy, w, h, cf;
      if (i < NYOLO) {
        cx = yolo[i];
        cy = yolo[NYOLO + i];
        w  = yolo[2 * NYOLO + i];
        h  = yolo[3 * NYOLO + i];
        cf = yolo[4 * NYOLO + i];
      } else {
        const float* p = rt + (i - NYOLO) * 5;
        cx = p[0]; cy = p[1]; w = p[2]; h = p[3];
        cf = p[4] / rmax;
      }
      const float hx = w * 0.5f, hy = h * 0.5f;
      const float a1 = cx - hx, b1 = cy - hy, a2 = cx + hx, b2 = cy + hy;
      x1[i] = a1; y1[i] = b1; x2[i] = a2; y2[i] = b2;
      sc[i] = cf;
      ky[i] = (cf >= 0.5f) ? cf : -1.0f;
      ar[i] = (a2 - a1) * (b2 - b1);
      if (cf >= 0.5f) cnt++;
    } else {
      ky[i] = -2.0f;                       // sorts strictly after everything
      sbox[i] = make_float4(0.f, 0.f, 0.f, 0.f);  // pads of sorted arrays
      ssc[i] = 0.f;
      sar[i] = 0.f;
    }
  }
  atomicAdd(&scount, cnt);
  __syncthreads();
  if (tid == 0) ((int*)(ws + 13 * NPAD))[0] = scount;
}

// ============ Kernel 2: stable descending rank + scatter ============
__global__ __launch_bounds__(256) void rank_scatter_kernel(float* __restrict__ ws) {
  __shared__ float skey[NPAD];
  const float* gkey = ws + 5 * NPAD;

  // Stage the 34.8 KB key array into LDS with CDNA5 async global->LDS copies
  // (global_load_async_to_lds_b128, tracked with ASYNCcnt), bypassing VGPRs.
  const uint32_t ldsbase = (uint32_t)(uintptr_t)(&skey[0]);
  for (int q = threadIdx.x; q < NPAD / 4; q += 256) {
    uint32_t loff = ldsbase + (uint32_t)q * 16u;
    const float* g = gkey + q * 4;
    asm volatile("global_load_async_to_lds_b128 %0, %1, off"
                 :: "v"(loff), "v"(g) : "memory");
  }
  asm volatile("s_wait_asynccnt 0" ::: "memory");
  __syncthreads();

  const int i = blockIdx.x * 256 + threadIdx.x;
  if (i >= NBOX) return;
  const float ki = skey[i];
  int cnt = 0;
  #pragma unroll 4
  for (int j = 0; j < NPAD; ++j) {
    const float kj = skey[j];
    cnt += (int)((kj > ki) | ((kj == ki) & (j < i)));
  }
  // scatter into sorted position
  float4* sbox = (float4*)(ws + 7 * NPAD);
  sbox[cnt] = make_float4(ws[i], ws[NPAD + i], ws[2 * NPAD + i], ws[3 * NPAD + i]);
  (ws + 11 * NPAD)[cnt] = ws[4 * NPAD + i];   // score
  (ws + 12 * NPAD)[cnt] = ws[6 * NPAD + i];   // area
}

// ============ Kernel 3: tiled greedy NMS + output (single block) ============
__global__ __launch_bounds__(1024) void nms_out_kernel(const float* __restrict__ ws,
                                                       float* __restrict__ out) {
  const float4* sbox = (const float4*)(ws + 7 * NPAD);
  const float* ssc = ws + 11 * NPAD;
  const float* sar = ws + 12 * NPAD;
  const int M = ((const int*)(ws + 13 * NPAD))[0];

  __shared__ unsigned char keep[NPAD];
  const int tid = threadIdx.x;
  for (int i = tid; i < NPAD; i += 1024) keep[i] = (i < M) ? 1 : 0;
  __syncthreads();

  const int Mt = (M + 15) & ~15;
  const int nTiles = Mt >> 4;
  const int wave = tid >> 5;
  const int lane = tid & 31;
  const bool lo = lane < 16;
  const int nl = lane & 15;

  for (int it = 0; it < nTiles; ++it) {
    const int ibase = it << 4;

    // uniform deadness check (all threads read same bytes -> uniform branch)
    unsigned alive = 0;
    #pragma unroll
    for (int a = 0; a < 16; ++a) alive |= keep[ibase + a];
    if (!alive) continue;

    // serial resolve of the 16x16 diagonal tile (exact greedy order)
    if (tid == 0) {
      for (int a = 0; a < 16; ++a) {
        const int i = ibase + a;
        if (!keep[i]) continue;
        const float4 bi = sbox[i];
        const float ai = sar[i];
        for (int b = a + 1; b < 16; ++b) {
          const int j = ibase + b;
          if (!keep[j]) continue;
          const float4 bj = sbox[j];
          float iw = fminf(bi.z, bj.z) - fmaxf(bi.x, bj.x);
          float ih = fminf(bi.w, bj.w) - fmaxf(bi.y, bj.y);
          iw = fmaxf(iw, 0.f); ih = fmaxf(ih, 0.f);
          const float inter = iw * ih;
          const float un = fmaxf(ai + sar[j] - inter, 1e-12f);
          if (inter + inter > un) keep[j] = 0;   // iou > 0.5
        }
      }
    }
    __syncthreads();

    // per-lane preload of this tile's rows (lanes 0-15: rows 0-7, 16-31: rows 8-15,
    // matching the 16x16 f32 WMMA C/D VGPR layout)
    float xi1[8], yi1[8], xi2[8], yi2[8];
    bool kb[8];
    const int rbase = ibase + (lo ? 0 : 8);
    #pragma unroll
    for (int m = 0; m < 8; ++m) {
      const float4 b = sbox[rbase + m];
      xi1[m] = b.x; yi1[m] = b.y; xi2[m] = b.z; yi2[m] = b.w;
      kb[m] = keep[rbase + m] != 0;
    }

    // A operand of V_WMMA_F32_16X16X4_F32: row M -> (area_i[M], 1, 0, 0)
    // lanes 0-15 hold K=0,1 of row M=lane; lanes 16-31 hold K=2,3 (zeros)
    v2f Aop;
    Aop.x = lo ? sar[ibase + nl] : 0.f;
    Aop.y = lo ? 1.f : 0.f;

    // parallel sweep of all later column tiles, one 16x16 tile per wave
    for (int jt = it + 1 + wave; jt < nTiles; jt += 32) {
      const int j = (jt << 4) + nl;          // this lane's column
      const float4 bj = sbox[j];
      const float aj = sar[j];
      // B operand: col N -> (1, area_j[N], 0, 0)^T
      v2f Bop;
      Bop.x = lo ? 1.f : 0.f;
      Bop.y = lo ? aj : 0.f;
      v8f c = {};
      // S[M][N] = area_i[M] + area_j[N] on the matrix pipe
      v8f S = __builtin_amdgcn_wmma_f32_16x16x4_f32(
          /*neg_a=*/false, Aop, /*neg_b=*/false, Bop,
          /*c_mod=*/(short)0, c, /*reuse_a=*/false, /*reuse_b=*/false);
      bool sup = false;
      #pragma unroll
      for (int m = 0; m < 8; ++m) {
        float iw = fminf(xi2[m], bj.z) - fmaxf(xi1[m], bj.x);
        float ih = fminf(yi2[m], bj.w) - fmaxf(yi1[m], bj.y);
        iw = fmaxf(iw, 0.f); ih = fmaxf(ih, 0.f);
        const float inter = iw * ih;
        const float un = fmaxf(S[m] - inter, 1e-12f);
        sup = sup || (kb[m] && (inter + inter > un));
      }
      if (sup) keep[j] = 0;   // idempotent byte store (both halves write 0)
    }
    __syncthreads();
  }

  // emit output: rows zeroed unless kept
  for (int i = tid; i < NBOX; i += 1024) {
    const float k = keep[i] ? 1.f : 0.f;
    const float4 b = sbox[i];
    out[i * 5 + 0] = b.x * k;
    out[i * 5 + 1] = b.y * k;
    out[i * 5 + 2] = b.z * k;
    out[i * 5 + 3] = b.w * k;
    out[i * 5 + 4] = ssc[i] * k;
  }
}

// ============ host launcher ============
extern "C" void kernel_launch(void* const* d_in, const int* in_sizes, int n_in,
                              void* d_out, int out_size, void* d_ws, size_t ws_size,
                              hipStream_t stream) {
  const float* yolo = (const float*)d_in[0];   // (1,5,8400)
  const float* rt   = (const float*)d_in[1];   // (1,300,5)
  float* ws  = (float*)d_ws;                   // needs ~452 KB
  float* out = (float*)d_out;                  // 8700*5 floats

  hipLaunchKernelGGL(prep_kernel, dim3(1), dim3(1024), 0, stream, yolo, rt, ws);
  hipLaunchKernelGGL(rank_scatter_kernel, dim3((NPAD + 255) / 256), dim3(256), 0, stream, ws);
  hipLaunchKernelGGL(nms_out_kernel, dim3(1), dim3(1024), 0, stream, ws, out);
}